// EgoStarSTGAT_45226005627088
// MI455X (gfx1250) — compile-verified
//
#include <hip/hip_runtime.h>
#include <hip/hip_bf16.h>

typedef float v2f __attribute__((ext_vector_type(2)));
typedef float v8f __attribute__((ext_vector_type(8)));
typedef int   vsi4 __attribute__((vector_size(16)));   // int4, matches builtin

#define B_   32
#define A_   50
#define T_   50
#define HID_ 128
#define H_   32
#define C_   16
#define HC_  512
#define R_   2500   // T_*A_
#define S_   2450   // T_*(A_-1)
#define RT_  157    // ceil(R_/16)
#define NEG_ 0.2f
#define CH_  128    // s-chunk staged in LDS for attention GEMM
#define NCH_ 20     // ceil(S_/CH_)

#if __has_builtin(__builtin_amdgcn_global_load_async_to_lds_b128) && \
    __has_builtin(__builtin_amdgcn_global_load_async_to_lds_b32)
#define HAVE_ASYNC_LDS 1
#endif

#define AS1_ __attribute__((address_space(1)))
#define AS3_ __attribute__((address_space(3)))

__device__ __forceinline__ float lrelu(float x) {
    return fmaxf(x, 0.f) + NEG_ * fminf(x, 0.f);
}

// ---------------------------------------------------------------------------
// Kernel 1: fold attention vectors into W:
//   wsrc[h][d] = sum_c W[h*C+c][d] * att_src[h][c]   (same for dst)
// ---------------------------------------------------------------------------
__global__ void k_attw(const float* __restrict__ W,
                       const float* __restrict__ avsrc,
                       const float* __restrict__ avdst,
                       float* __restrict__ wsrc, float* __restrict__ wdst) {
    int i = blockIdx.x * blockDim.x + threadIdx.x;
    if (i >= H_ * HID_) return;
    int h = i / HID_, d = i % HID_;
    float ss = 0.f, sd = 0.f;
    for (int c = 0; c < C_; ++c) {
        float w = W[(h * C_ + c) * HID_ + d];
        ss += w * avsrc[h * C_ + c];
        sd += w * avdst[h * C_ + c];
    }
    wsrc[i] = ss;
    wdst[i] = sd;
}

// ---------------------------------------------------------------------------
// Bias broadcast over the whole output; ego slots overwritten later by k_attn.
// ---------------------------------------------------------------------------
__global__ void k_bias(const float* __restrict__ bias, float* __restrict__ out,
                       long n) {
    long i = (long)blockIdx.x * blockDim.x + threadIdx.x;
    long stride = (long)gridDim.x * blockDim.x;
    for (; i < n; i += stride) out[i] = bias[i & (HC_ - 1)];
}

// ---------------------------------------------------------------------------
// Kernel 2: projection GEMM  xw[b,h,s,c] = sum_d x[b,r,d] * W[h*16+c][d]
// via V_WMMA_F32_16X16X4_F32, plus fused score GEMM on waves 0..3:
//   a_src[b,h,s] = x . wsrc[h]   (N-cols = heads),  a_dst at ego rows.
// x tile staged once in LDS (pitch 132, conflict-free); 8 waves x 4 heads.
// Non-ego rows re-indexed by s = r - t - 1, head-major layout.
// ---------------------------------------------------------------------------
__global__ __launch_bounds__(256) void k_xw(const float* __restrict__ hin,
                                            const float* __restrict__ Wd,
                                            const float* __restrict__ wsrc,
                                            const float* __restrict__ wdst,
                                            float* __restrict__ xw,
                                            float* __restrict__ asrc,
                                            float* __restrict__ adst) {
    __shared__ float xa[16 * 132];
    int blk = blockIdx.x;
    int b = blk / RT_, rt = blk % RT_;
    int r0 = rt * 16;
    int tid = threadIdx.x;

    for (int i = tid; i < 16 * HID_; i += 256) {
        int m = i >> 7, d = i & 127;
        int r = r0 + m;
        float v = 0.f;
        if (r < R_) {
            int t = r / A_, a = r % A_;
            v = hin[(((size_t)b * A_ + a) * T_ + t) * HID_ + d];
        }
        xa[m * 132 + d] = v;
    }
    __syncthreads();

    int wave = tid >> 5, lane = tid & 31;
    int mrow = lane & 15;               // A/B lane row, D column
    int kp = (lane < 16) ? 0 : 2;       // K-pair offset per 16x4 layout
    int half = (lane >> 4) << 3;        // D rows: j (+8 for upper half-wave)

    // Pre-decode per-output-row mapping once (shared by all heads).
    int sJ[8], tJ[8];
#pragma unroll
    for (int j = 0; j < 8; ++j) {
        int r = r0 + j + half;
        sJ[j] = -1; tJ[j] = -1;
        if (r < R_) {
            int t = r / A_, a = r % A_;
            if (a > 0) sJ[j] = r - t - 1;   // non-ego -> s index
            else       tJ[j] = t;           // ego row -> t index
        }
    }

    for (int hh = 0; hh < 4; ++hh) {
        int head = wave + hh * 8;
        const float* wrow = Wd + ((size_t)(head * 16 + mrow)) * HID_;
        v8f acc = {};
#pragma unroll 8
        for (int k = 0; k < 32; ++k) {
            int d0 = k * 4 + kp;
            v2f av = { xa[mrow * 132 + d0], xa[mrow * 132 + d0 + 1] };
            v2f bv = { wrow[d0], wrow[d0 + 1] };
            acc = __builtin_amdgcn_wmma_f32_16x16x4_f32(
                false, av, false, bv, (short)0, acc, false, false);
        }
        size_t base = ((size_t)b * H_ + head) * S_ * C_ + mrow;
#pragma unroll
        for (int j = 0; j < 8; ++j) {
            if (sJ[j] >= 0) xw[base + (size_t)sJ[j] * C_] = acc[j];
        }
    }

    // Fused score GEMM: waves 0..3, N = 16 heads each.
    //   wave 0/1 -> a_src heads 0-15 / 16-31 ; wave 2/3 -> a_dst heads.
    if (wave < 4) {
        int isDst = wave >> 1;
        int hbase = (wave & 1) * 16;
        const float* wf = (isDst ? wdst : wsrc) +
                          ((size_t)(hbase + mrow)) * HID_;
        v8f acc = {};
#pragma unroll 8
        for (int k = 0; k < 32; ++k) {
            int d0 = k * 4 + kp;
            v2f av = { xa[mrow * 132 + d0], xa[mrow * 132 + d0 + 1] };
            v2f bv = { wf[d0], wf[d0 + 1] };
            acc = __builtin_amdgcn_wmma_f32_16x16x4_f32(
                false, av, false, bv, (short)0, acc, false, false);
        }
        int head = hbase + mrow;        // D column = head
        if (!isDst) {
            size_t base = ((size_t)b * H_ + head) * S_;
#pragma unroll
            for (int j = 0; j < 8; ++j)
                if (sJ[j] >= 0) asrc[base + sJ[j]] = acc[j];
        } else {
            size_t base = ((size_t)b * H_ + head) * T_;
#pragma unroll
            for (int j = 0; j < 8; ++j)
                if (tJ[j] >= 0) adst[base + tJ[j]] = acc[j];
        }
    }
}

// ---------------------------------------------------------------------------
// Kernel 4: softmax stats per (b,h,t): row max and 1/denominator over S.
// a_src slab (9.8 KB) staged in LDS; wave-level shfl reductions (wave32).
// ---------------------------------------------------------------------------
__global__ __launch_bounds__(256) void k_stats(const float* __restrict__ asrc,
                                               const float* __restrict__ adst,
                                               float* __restrict__ mw,
                                               float* __restrict__ rdw) {
    __shared__ float sa[S_];
    int bh = blockIdx.x;
    int tid = threadIdx.x;
    const float* src = &asrc[(size_t)bh * S_];
    for (int i = tid; i < S_; i += 256) sa[i] = src[i];
    __syncthreads();
    int wave = tid >> 5, lane = tid & 31;
    for (int t = wave; t < T_; t += 8) {
        float ad = adst[bh * T_ + t];
        float mx = -3.0e38f;
        for (int s = lane; s < S_; s += 32)
            mx = fmaxf(mx, lrelu(sa[s] + ad));
        for (int off = 16; off > 0; off >>= 1)
            mx = fmaxf(mx, __shfl_xor(mx, off, 32));
        float sm = 0.f;
        for (int s = lane; s < S_; s += 32)
            sm += __expf(lrelu(sa[s] + ad) - mx);
        for (int off = 16; off > 0; off >>= 1)
            sm += __shfl_xor(sm, off, 32);
        if (lane == 0) {
            mw[bh * T_ + t] = mx;
            rdw[bh * T_ + t] = 1.0f / sm;
        }
    }
}

// ---------------------------------------------------------------------------
// Kernel 5: out_ego[b,t,h,c] = sum_s att[b,t,s,h] * xw[b,h,s,c]
// One workgroup per (b,h); 4 waves cover t-tiles 0..63 (mask t>=50).
// att generated in-register inside the WMMA A-operand: exp(lrelu(.)-m)*rd.
// V tiles streamed through LDS in 128-row chunks via async global->LDS b128
// (ASYNCcnt + s_wait_asynccnt) when available; ragged tail uses guarded
// scalar loads so padded K-steps see zeros. K = 2450 -> 640 WMMAs/wave.
// ---------------------------------------------------------------------------
__global__ __launch_bounds__(128) void k_attn(const float* __restrict__ xw,
        const float* __restrict__ asrc, const float* __restrict__ adst,
        const float* __restrict__ mw, const float* __restrict__ rdw,
        const float* __restrict__ bias, float* __restrict__ out) {
    __shared__ __align__(16) float xv[CH_ * C_];
    __shared__ __align__(16) float sas[CH_];
    int bh = blockIdx.x;
    int b = bh >> 5, head = bh & 31;
    int tid = threadIdx.x;
    int wave = tid >> 5, lane = tid & 31;
    int t0 = wave * 16;
    int tt = t0 + (lane & 15);          // A-operand row for this lane
    float ad = 0.f, mm = 3.0e38f, rr = 0.f;  // invalid rows -> att == 0
    if (tt < T_) {
        ad = adst[bh * T_ + tt];
        mm = mw[bh * T_ + tt];
        rr = rdw[bh * T_ + tt];
    }
    int kp = (lane < 16) ? 0 : 2;
    int cc = lane & 15;
    v8f acc = {};

    for (int ch = 0; ch < NCH_; ++ch) {
        int sbase = ch * CH_;
        size_t gb = ((size_t)bh * S_ + sbase) * C_;
#ifdef HAVE_ASYNC_LDS
        if (sbase + CH_ <= S_) {
            // Full chunk: async DMA global -> LDS, no VGPR round trip.
            const float* gsrc = xw + gb;
#pragma unroll
            for (int u = 0; u < 4; ++u) {
                int foff = tid * 4 + u * 512;      // float index, 16B aligned
                __builtin_amdgcn_global_load_async_to_lds_b128(
                    (AS1_ vsi4*)(uintptr_t)(gsrc + foff),
                    (AS3_ vsi4*)(uintptr_t)&xv[foff], 0, 0);
            }
            __builtin_amdgcn_global_load_async_to_lds_b32(
                (AS1_ int*)(uintptr_t)(asrc + (size_t)bh * S_ + sbase + tid),
                (AS3_ int*)(uintptr_t)&sas[tid], 0, 0);
#if __has_builtin(__builtin_amdgcn_s_wait_asynccnt)
            __builtin_amdgcn_s_wait_asynccnt(0);
#else
            asm volatile("s_wait_asynccnt 0x0" ::: "memory");
#endif
        } else
#endif
        {
            for (int i = tid; i < CH_ * C_; i += 128) {
                int sl = i >> 4;
                xv[i] = (sbase + sl < S_) ? xw[gb + i] : 0.f;
            }
            sas[tid] = (sbase + tid < S_) ? asrc[(size_t)bh * S_ + sbase + tid]
                                          : 0.f;
        }
        __syncthreads();
#pragma unroll 8
        for (int k = 0; k < 32; ++k) {
            int sl = k * 4 + kp;
            float a0 = sas[sl], a1 = sas[sl + 1];
            float att0 = __expf(lrelu(a0 + ad) - mm) * rr;
            float att1 = __expf(lrelu(a1 + ad) - mm) * rr;
            v2f av = { att0, att1 };
            v2f bv = { xv[sl * C_ + cc], xv[(sl + 1) * C_ + cc] };
            acc = __builtin_amdgcn_wmma_f32_16x16x4_f32(
                false, av, false, bv, (short)0, acc, false, false);
        }
        __syncthreads();
    }

    int half = (lane >> 4) << 3;
    float bb = bias[head * C_ + cc];
#pragma unroll
    for (int j = 0; j < 8; ++j) {
        int t = t0 + j + half;          // D-row mapping (VGPR j, half-wave)
        if (t < T_) {
            out[(((size_t)b * A_ + 0) * T_ + t) * HC_ + head * C_ + cc] =
                acc[j] + bb;
        }
    }
}

// ---------------------------------------------------------------------------
extern "C" void kernel_launch(void* const* d_in, const int* in_sizes, int n_in,
                              void* d_out, int out_size, void* d_ws,
                              size_t ws_size, hipStream_t stream) {
    (void)in_sizes; (void)n_in; (void)out_size; (void)ws_size;
    const float* hin  = (const float*)d_in[0];   // (B,A,T,HID)
    const float* W    = (const float*)d_in[1];   // (H*C,HID)
    const float* avs  = (const float*)d_in[2];   // (H,C)
    const float* avd  = (const float*)d_in[3];   // (H,C)
    const float* bias = (const float*)d_in[4];   // (H*C)
    float* out = (float*)d_out;                  // (B,A,T,H*C)

    float* ws = (float*)d_ws;
    size_t off = 0;
    float* xw   = ws + off; off += (size_t)B_ * H_ * S_ * C_;  // 153 MB
    float* asrc = ws + off; off += (size_t)B_ * H_ * S_;       // 10 MB
    float* adst = ws + off; off += (size_t)B_ * H_ * T_;
    float* mw   = ws + off; off += (size_t)B_ * H_ * T_;
    float* rdw  = ws + off; off += (size_t)B_ * H_ * T_;
    float* wsrc = ws + off; off += (size_t)H_ * HID_;
    float* wdst = ws + off; off += (size_t)H_ * HID_;

    k_attw<<<16, 256, 0, stream>>>(W, avs, avd, wsrc, wdst);

    long n_out = (long)B_ * A_ * T_ * HC_;
    k_bias<<<2048, 256, 0, stream>>>(bias, out, n_out);

    k_xw<<<B_ * RT_, 256, 0, stream>>>(hin, W, wsrc, wdst, xw, asrc, adst);

    k_stats<<<B_ * H_, 256, 0, stream>>>(asrc, adst, mw, rdw);

    k_attn<<<B_ * H_, 128, 0, stream>>>(xw, asrc, adst, mw, rdw, bias, out);
}